// FedAux_69475390980138
// MI455X (gfx1250) — compile-verified
//
#include <hip/hip_runtime.h>
#include <stdint.h>

#define N_NODES 16384
#define N_EDGES 262144
#define F_DIM   128
#define N_CLS   10

typedef float v2f __attribute__((ext_vector_type(2)));
typedef float v8f __attribute__((ext_vector_type(8)));

// ---------------------------------------------------------------------------
// f32 WMMA: D[16x16] += A[16x4] * B[4x16]  (wave32, CDNA5 V_WMMA_F32_16X16X4_F32)
// A layout: lane<16 -> row=lane&15, k in {0,1}; lane>=16 -> k in {2,3}
// B layout: lane<16 -> col=lane&15, k in {0,1}; lane>=16 -> k in {2,3}
// C/D layout: vgpr v, lanes 0-15 -> M=v, N=lane; lanes 16-31 -> M=v+8, N=lane-16
// ---------------------------------------------------------------------------
__device__ __forceinline__ v8f wmma_f32_16x16x4(v2f a, v2f b, v8f c) {
#if defined(__gfx1250__) && __has_builtin(__builtin_amdgcn_wmma_f32_16x16x4_f32)
    return __builtin_amdgcn_wmma_f32_16x16x4_f32(false, a, false, b, (short)0, c,
                                                 false, false);
#else
    (void)a; (void)b;
    return c;   // host pass / non-gfx1250: never executed
#endif
}

// ---------------------------------------------------------------------------
// Tensor Data Mover: DMA `nrows` rows of 128 f32 from global into LDS with a
// hardware-padded pitch of 136 dwords (pad 8 dwords after every 128).
// D# per CDNA5 ISA ch.8: group0 = {count, lds_addr, global_addr, type=2};
// group1 = {data_size=4B, pad_enable, pad_interval=128dw, pad_amount=8dw,
//           tensor_dim0=tile_dim0=128, tensor_dim1=tile_dim1=nrows,
//           tensor_dim0_stride=128}; groups 2/3 unused (2D tile).
// This toolchain's builtin is the 6-arg form (g0, g1, g2, g3, g4, cpol);
// trailing groups zero-filled.
// ---------------------------------------------------------------------------
#if defined(__gfx1250__) && __has_builtin(__builtin_amdgcn_tensor_load_to_lds) && \
    __has_builtin(__builtin_amdgcn_s_wait_tensorcnt)
#define HAVE_TDM 1
typedef unsigned int u32x4 __attribute__((ext_vector_type(4)));
typedef int          i32x4 __attribute__((ext_vector_type(4)));
typedef int          i32x8 __attribute__((ext_vector_type(8)));

__device__ __forceinline__ void tdm_load_rows(const float* lds_dst,
                                              const float* gsrc, int nrows) {
    const uint64_t ga = (uint64_t)(uintptr_t)gsrc;
    const uint32_t la = (uint32_t)(uintptr_t)lds_dst;   // low 32 bits = LDS byte addr
    u32x4 g0;
    g0[0] = 1u;                                     // count = 1 (valid user D#)
    g0[1] = la;                                     // lds_addr (bytes)
    g0[2] = (uint32_t)ga;                           // global_addr[31:0]
    g0[3] = (uint32_t)((ga >> 32) & 0x01FFFFFFu)    // global_addr[56:32]
          | (2u << 30);                             // type = 2
    i32x8 g1;
    g1[0] = (int)((2u << 16)       // data_size = 4 bytes
                | (1u << 20)       // pad_enable
                | (6u << 22)       // pad_interval code 6 -> 128 dwords (one row)
                | (7u << 25));     // pad_amount  code 7 -> 8 dwords (pitch 136)
    g1[1] = (int)(128u << 16);     // tensor_dim0 = 128 elements
    g1[2] = (int)((uint32_t)nrows << 16);  // tensor_dim1 = nrows
    g1[3] = (int)(128u << 16);     // tile_dim0 = 128
    g1[4] = nrows;                 // tile_dim1 = nrows (tile_dim2 = 0)
    g1[5] = 128;                   // tensor_dim0_stride = 128 elements
    g1[6] = 0;                     // tensor_dim1_stride unused (2D)
    g1[7] = 0;
    i32x4 gz4 = {0, 0, 0, 0};
    i32x8 gz8 = {0, 0, 0, 0, 0, 0, 0, 0};
    __builtin_amdgcn_tensor_load_to_lds(g0, g1, gz4, gz4, gz8, 0);
}
#else
#define HAVE_TDM 0
#endif

// ---------------------------------------------------------------------------
__global__ void k_zero(float* __restrict__ p, int n) {
    int i = blockIdx.x * blockDim.x + threadIdx.x;
    if (i < n) p[i] = 0.0f;
}

__global__ void k_deg(const int* __restrict__ dst, float* __restrict__ deg) {
    int e = blockIdx.x * blockDim.x + threadIdx.x;
    if (e < N_EDGES) atomicAdd(&deg[dst[e]], 1.0f);
}

// one wave per edge; each lane moves 4 contiguous features
__global__ void k_scatter(const float* __restrict__ x, const int* __restrict__ src,
                          const int* __restrict__ dst, float* __restrict__ agg) {
    int gid  = blockIdx.x * blockDim.x + threadIdx.x;
    int e    = gid >> 5;
    int lane = gid & 31;
    if (e >= N_EDGES) return;
    int s = src[e], d = dst[e];
    const float4 v = *reinterpret_cast<const float4*>(x + (size_t)s * F_DIM + lane * 4);
    float* ap = agg + (size_t)d * F_DIM + lane * 4;
    atomicAdd(ap + 0, v.x);
    atomicAdd(ap + 1, v.y);
    atomicAdd(ap + 2, v.z);
    atomicAdd(ap + 3, v.w);
}

// ---------------------------------------------------------------------------
// out = act( (agg/max(deg,1)) @ Wl^T + x @ Wr^T + bl )
// grid: N/16 blocks x 256 threads (8 waves); wave w -> 16x16 tile (rows rb, cols w*16)
// ---------------------------------------------------------------------------
__global__ __launch_bounds__(256) void k_sage(
        const float* __restrict__ xin, const float* __restrict__ agg,
        const float* __restrict__ deg, const float* __restrict__ Wl,
        const float* __restrict__ bl,  const float* __restrict__ Wr,
        float* __restrict__ out, int do_relu) {
    const int lane = threadIdx.x & 31;
    const int ct   = threadIdx.x >> 5;       // column tile 0..7
    const int rb   = blockIdx.x * 16;        // row base
    const int r    = lane & 15;
    const int koff = (lane >> 4) * 2;        // K split between lane halves
    const int row  = rb + r;
    const int col  = ct * 16 + r;

    const float invd = 1.0f / fmaxf(deg[row], 1.0f);

    v8f acc = {};
    #pragma unroll
    for (int k = 0; k < F_DIM; k += 4) {
        v2f a, b;
        // neighbor-mean path:  A = agg/deg,  B[k][n] = Wl[n][k]
        a[0] = agg[(size_t)row * F_DIM + k + koff]     * invd;
        a[1] = agg[(size_t)row * F_DIM + k + koff + 1] * invd;
        b[0] = Wl[(size_t)col * F_DIM + k + koff];
        b[1] = Wl[(size_t)col * F_DIM + k + koff + 1];
        acc = wmma_f32_16x16x4(a, b, acc);
        // root path:  A = x,  B[k][n] = Wr[n][k]
        a[0] = xin[(size_t)row * F_DIM + k + koff];
        a[1] = xin[(size_t)row * F_DIM + k + koff + 1];
        b[0] = Wr[(size_t)col * F_DIM + k + koff];
        b[1] = Wr[(size_t)col * F_DIM + k + koff + 1];
        acc = wmma_f32_16x16x4(a, b, acc);
    }
    const float bias = bl[col];
    #pragma unroll
    for (int v = 0; v < 8; ++v) {
        int m = v + ((lane >> 4) << 3);
        float val = acc[v] + bias;
        if (do_relu) val = fmaxf(val, 0.0f);
        out[(size_t)(rb + m) * F_DIM + ct * 16 + r] = val;
    }
}

// ---------------------------------------------------------------------------
// score[i] = (h_i . aux) / (max(|h_i|,eps) * max(|aux|,eps)); one wave per row
// ---------------------------------------------------------------------------
__global__ __launch_bounds__(256) void k_score(const float* __restrict__ h,
                                               const float* __restrict__ aux,
                                               float* __restrict__ score) {
    const int lane = threadIdx.x & 31;
    const int i    = blockIdx.x * 8 + (threadIdx.x >> 5);
    float dot = 0.f, nh = 0.f, na = 0.f;
    #pragma unroll
    for (int j = 0; j < 4; ++j) {
        int d    = lane * 4 + j;
        float hv = h[(size_t)i * F_DIM + d];
        float av = aux[d];
        dot += hv * av; nh += hv * hv; na += av * av;
    }
    #pragma unroll
    for (int off = 16; off > 0; off >>= 1) {
        dot += __shfl_xor(dot, off);
        nh  += __shfl_xor(nh,  off);
        na  += __shfl_xor(na,  off);
    }
    if (lane == 0) {
        const float eps = 1e-8f;
        score[i] = dot / (fmaxf(sqrtf(nh), eps) * fmaxf(sqrtf(na), eps));
    }
}

// ---------------------------------------------------------------------------
// z = (kappa @ h) / rowsum(kappa),  kappa_ij = exp(-(s_j - s_i)^2)   [SIGMA=1]
// grid: N/64 blocks x 128 threads (4 waves); wave -> 16 output rows, full 128 cols
// j-tile of h staged in LDS, TDM double-buffered (pitch 136 via TDM pad ->
// lane halves hit disjoint bank ranges)
// ---------------------------------------------------------------------------
#define SH_PITCH 136

// one 16x16 kappa tile: 8 exps/lane -> 4 A-fragments, 32 WMMAs against LDS tile
__device__ __forceinline__ void kappa_tile_mac(const float* __restrict__ tile,
                                               const float* __restrict__ sj,
                                               float s_i, int r, int koff,
                                               float& rsum, v8f acc[8]) {
    v2f af[4];
    #pragma unroll
    for (int t = 0; t < 4; ++t) {
        float d0 = sj[t * 4 + koff]     - s_i;
        float d1 = sj[t * 4 + koff + 1] - s_i;
        float e0 = __expf(-d0 * d0);
        float e1 = __expf(-d1 * d1);
        af[t][0] = e0; af[t][1] = e1;
        rsum += e0 + e1;
    }
    #pragma unroll
    for (int ct = 0; ct < 8; ++ct) {
        #pragma unroll
        for (int t = 0; t < 4; ++t) {
            v2f b;
            b[0] = tile[(t * 4 + koff)     * SH_PITCH + ct * 16 + r];
            b[1] = tile[(t * 4 + koff + 1) * SH_PITCH + ct * 16 + r];
            acc[ct] = wmma_f32_16x16x4(af[t], b, acc[ct]);
        }
    }
}

__global__ __launch_bounds__(128) void k_kappa(const float* __restrict__ h,
                                               const float* __restrict__ score,
                                               float* __restrict__ z) {
    __shared__ float shh[2][16 * SH_PITCH];
    const int tid  = threadIdx.x;
    const int lane = tid & 31;
    const int wave = tid >> 5;                      // 0..3
    const int rb   = (blockIdx.x * 4 + wave) * 16;  // this wave's 16 rows
    const int r    = lane & 15;
    const int koff = (lane >> 4) * 2;

    const float s_i = score[rb + r];
    float rsum = 0.0f;

    v8f acc[8];
    {
        v8f zero8 = {};
        #pragma unroll
        for (int t = 0; t < 8; ++t) acc[t] = zero8;
    }

#if HAVE_TDM
    // Each wave DMAs its own 4-row slice (no divergence: TDM ignores EXEC, so
    // all waves issuing disjoint slices keeps issue uniform and load balanced).
    tdm_load_rows(&shh[0][wave * 4 * SH_PITCH], h + (size_t)(wave * 4) * F_DIM, 4);
    __builtin_amdgcn_s_wait_tensorcnt(0);
    __syncthreads();
    for (int jb = 0; jb < N_NODES; jb += 16) {
        const int cur = (jb >> 4) & 1;
        if (jb + 16 < N_NODES)      // kick next tile while computing this one
            tdm_load_rows(&shh[cur ^ 1][wave * 4 * SH_PITCH],
                          h + (size_t)(jb + 16 + wave * 4) * F_DIM, 4);
        kappa_tile_mac(&shh[cur][0], score + jb, s_i, r, koff, rsum, acc);
        __builtin_amdgcn_s_wait_tensorcnt(0);   // next buffer resident
        __syncthreads();                        // all waves done reading + loaded
    }
#else
    for (int jb = 0; jb < N_NODES; jb += 16) {
        #pragma unroll
        for (int t = 0; t < 4; ++t) {
            int idx = tid * 4 + t;        // 0..511 float4 slots
            int rr  = idx >> 5;           // 0..15
            int cc  = (idx & 31) * 4;     // 0..124
            *reinterpret_cast<float4*>(&shh[0][rr * SH_PITCH + cc]) =
                *reinterpret_cast<const float4*>(&h[(size_t)(jb + rr) * F_DIM + cc]);
        }
        if (jb + 16 < N_NODES)
            __builtin_prefetch(&h[(size_t)(jb + 16) * F_DIM + tid * 16], 0, 0);
        __syncthreads();
        kappa_tile_mac(&shh[0][0], score + jb, s_i, r, koff, rsum, acc);
        __syncthreads();
    }
#endif

    // complete row sums (lane L holds k%4 in {0,1}, lane L+16 holds {2,3})
    float tot = rsum + __shfl_xor(rsum, 16);
    #pragma unroll
    for (int ct = 0; ct < 8; ++ct) {
        #pragma unroll
        for (int v = 0; v < 8; ++v) {
            int m    = v + ((lane >> 4) << 3);
            float rs = __shfl(tot, m);    // lane m (<16) holds rowsum of row m
            z[(size_t)(rb + m) * F_DIM + ct * 16 + r] = acc[ct][v] / rs;
        }
    }
}

// ---------------------------------------------------------------------------
// out[i][c] = bc[c] + h_i . Wc[c][:128] + z_i . Wc[c][128:]
// ---------------------------------------------------------------------------
__global__ void k_cls(const float* __restrict__ h, const float* __restrict__ z,
                      const float* __restrict__ Wc, const float* __restrict__ bc,
                      float* __restrict__ out) {
    int gid = blockIdx.x * blockDim.x + threadIdx.x;
    if (gid >= N_NODES * N_CLS) return;
    int i = gid / N_CLS, c = gid % N_CLS;
    const float* hr = h + (size_t)i * F_DIM;
    const float* zr = z + (size_t)i * F_DIM;
    const float* w1 = Wc + (size_t)c * (2 * F_DIM);
    const float* w2 = w1 + F_DIM;
    float accv = bc[c];
    #pragma unroll 8
    for (int d = 0; d < F_DIM; ++d)
        accv += hr[d] * w1[d] + zr[d] * w2[d];
    out[gid] = accv;
}

// ---------------------------------------------------------------------------
extern "C" void kernel_launch(void* const* d_in, const int* in_sizes, int n_in,
                              void* d_out, int out_size, void* d_ws, size_t ws_size,
                              hipStream_t stream) {
    (void)in_sizes; (void)n_in; (void)out_size; (void)ws_size;
    const float* x   = (const float*)d_in[0];
    const int*   ei  = (const int*)  d_in[1];
    const float* W1l = (const float*)d_in[2];
    const float* b1l = (const float*)d_in[3];
    const float* W1r = (const float*)d_in[4];
    const float* W2l = (const float*)d_in[5];
    const float* b2l = (const float*)d_in[6];
    const float* W2r = (const float*)d_in[7];
    const float* aux = (const float*)d_in[8];
    const float* Wc  = (const float*)d_in[9];
    const float* bc  = (const float*)d_in[10];
    const int* src = ei;
    const int* dst = ei + N_EDGES;

    float* ws    = (float*)d_ws;
    float* agg   = ws;  ws += (size_t)N_NODES * F_DIM;   // 8 MB
    float* deg   = ws;  ws += N_NODES;                   // contiguous after agg
    float* h1    = ws;  ws += (size_t)N_NODES * F_DIM;
    float* h2    = ws;  ws += (size_t)N_NODES * F_DIM;
    float* score = ws;  ws += N_NODES;
    float* zbuf  = ws;  ws += (size_t)N_NODES * F_DIM;
    float* out   = (float*)d_out;

    // ---- SAGE layer 1 ----
    k_zero<<<(N_NODES * (F_DIM + 1) + 255) / 256, 256, 0, stream>>>(agg, N_NODES * (F_DIM + 1));
    k_deg<<<(N_EDGES + 255) / 256, 256, 0, stream>>>(dst, deg);
    k_scatter<<<(N_EDGES * 32) / 256, 256, 0, stream>>>(x, src, dst, agg);
    k_sage<<<N_NODES / 16, 256, 0, stream>>>(x, agg, deg, W1l, b1l, W1r, h1, 1);

    // ---- SAGE layer 2 (deg reused) ----
    k_zero<<<(N_NODES * F_DIM + 255) / 256, 256, 0, stream>>>(agg, N_NODES * F_DIM);
    k_scatter<<<(N_EDGES * 32) / 256, 256, 0, stream>>>(h1, src, dst, agg);
    k_sage<<<N_NODES / 16, 256, 0, stream>>>(h1, agg, deg, W2l, b2l, W2r, h2, 0);

    // ---- kernel aggregation (dominant stage) ----
    k_score<<<N_NODES / 8, 256, 0, stream>>>(h2, aux, score);
    k_kappa<<<N_NODES / 64, 128, 0, stream>>>(h2, score, zbuf);

    // ---- classifier ----
    k_cls<<<(N_NODES * N_CLS + 255) / 256, 256, 0, stream>>>(h2, zbuf, Wc, bc, out);
}